// DDE_72988674228564
// MI455X (gfx1250) — compile-verified
//
#include <hip/hip_runtime.h>

#define WAVE 32
#define BLOCK 256
#define WAVES_PER_BLOCK (BLOCK / WAVE)
#define K 4                              // b128 loads per lane per array per tile
#define TILE_EDGES (WAVE * 4 * K)        // 512 edges per wave-tile
#define SRC_BYTES (TILE_EDGES * 4)       // 2048 B
#define BUF_BYTES (2 * SRC_BYTES)        // src+dst region per buffer: 4096 B
#define WAVE_LDS (2 * BUF_BYTES)         // double buffered: 8192 B per wave
#define BLOCK_LDS (WAVES_PER_BLOCK * WAVE_LDS)  // 64 KB per block (LDS is 320KB/WGP)

// Scatter h[src] into acc[dst] with float atomics; optionally count degrees.
// Edge lists are streamed HBM -> LDS with CDNA5 async-to-LDS loads, double
// buffered per wave, synchronized with s_wait_asynccnt (in-order completion).
template <bool COUNT>
__global__ __launch_bounds__(BLOCK) void scatter_kernel(
    const int* __restrict__ src, const int* __restrict__ dst,
    const float* __restrict__ h, float* __restrict__ acc,
    float* __restrict__ cnt, int nEdges)
{
    __shared__ __align__(16) char smem[BLOCK_LDS];

    const int lane      = threadIdx.x & (WAVE - 1);
    const int waveInBlk = threadIdx.x >> 5;
    const int waveGlob  = blockIdx.x * WAVES_PER_BLOCK + waveInBlk;
    const int nWaves    = gridDim.x * WAVES_PER_BLOCK;
    const int nTiles    = (nEdges + TILE_EDGES - 1) / TILE_EDGES;
    const int maxElem   = nEdges - 4;              // last legal 4-edge group

    char* const waveLds = smem + waveInBlk * WAVE_LDS;
    const unsigned long long srcBase = (unsigned long long)src;
    const unsigned long long dstBase = (unsigned long long)dst;

    // Issue K async b128 loads of src and dst for one tile into buffer `buf`.
    auto issue_tile = [&](int tile, int buf) {
        const int e0 = tile * TILE_EDGES;
        char* bufS = waveLds + buf * BUF_BYTES;          // src region
        char* bufD = bufS + SRC_BYTES;                   // dst region
#pragma unroll
        for (int k = 0; k < K; ++k) {
            int elem = e0 + (k * WAVE + lane) * 4;
            if (elem > maxElem) elem = maxElem;          // clamp tail (guarded later)
            unsigned long long gaS = srcBase + (unsigned long long)elem * 4ull;
            unsigned long long gaD = dstBase + (unsigned long long)elem * 4ull;
            // low 32 bits of a flat shared pointer == LDS byte offset
            unsigned lsS = (unsigned)(unsigned long long)(bufS + (k * WAVE + lane) * 16);
            unsigned lsD = (unsigned)(unsigned long long)(bufD + (k * WAVE + lane) * 16);
            asm volatile("global_load_async_to_lds_b128 %0, %1, off"
                         :: "v"(lsS), "v"(gaS) : "memory");
            asm volatile("global_load_async_to_lds_b128 %0, %1, off"
                         :: "v"(lsD), "v"(gaD) : "memory");
        }
    };

    auto process_tile = [&](int tile, int buf) {
        const int e0 = tile * TILE_EDGES;
        const int4* sP = (const int4*)(waveLds + buf * BUF_BYTES);
        const int4* dP = (const int4*)(waveLds + buf * BUF_BYTES + SRC_BYTES);
#pragma unroll
        for (int k = 0; k < K; ++k) {
            int4 s = sP[k * WAVE + lane];                // ds_load_b128
            int4 d = dP[k * WAVE + lane];
            int e = e0 + (k * WAVE + lane) * 4;
            if (e + 3 < nEdges) {
                atomicAdd(&acc[d.x], h[s.x]);
                atomicAdd(&acc[d.y], h[s.y]);
                atomicAdd(&acc[d.z], h[s.z]);
                atomicAdd(&acc[d.w], h[s.w]);
                if (COUNT) {
                    atomicAdd(&cnt[d.x], 1.0f);
                    atomicAdd(&cnt[d.y], 1.0f);
                    atomicAdd(&cnt[d.z], 1.0f);
                    atomicAdd(&cnt[d.w], 1.0f);
                }
            } else {
                if (e + 0 < nEdges) { atomicAdd(&acc[d.x], h[s.x]); if (COUNT) atomicAdd(&cnt[d.x], 1.0f); }
                if (e + 1 < nEdges) { atomicAdd(&acc[d.y], h[s.y]); if (COUNT) atomicAdd(&cnt[d.y], 1.0f); }
                if (e + 2 < nEdges) { atomicAdd(&acc[d.z], h[s.z]); if (COUNT) atomicAdd(&cnt[d.z], 1.0f); }
                if (e + 3 < nEdges) { atomicAdd(&acc[d.w], h[s.w]); if (COUNT) atomicAdd(&cnt[d.w], 1.0f); }
            }
        }
    };

    int tile = waveGlob;
    if (tile >= nTiles) return;
    issue_tile(tile, 0);
    int buf = 0;
    for (; tile < nTiles; tile += nWaves) {
        const int next = tile + nWaves;
        if (next < nTiles) {
            issue_tile(next, buf ^ 1);
            // 8 newer ops outstanding; asynccnt<=8 => the 8 for `buf` completed
            asm volatile("s_wait_asynccnt 0x8" ::: "memory");
        } else {
            asm volatile("s_wait_asynccnt 0x0" ::: "memory");
        }
        process_tile(tile, buf);
        buf ^= 1;
    }
}

// out[i] = acc[i] / max(cnt[i], 1); also zeroes acc for the next round
// (fused zeroing replaces the inter-round memset pass). Vectorized x4.
__global__ __launch_bounds__(256) void norm_kernel(
    float* __restrict__ acc, const float* __restrict__ cnt,
    float* __restrict__ out, int n)
{
    int i = (blockIdx.x * blockDim.x + threadIdx.x) * 4;
    if (i + 3 < n) {
        float4 a = *(const float4*)(acc + i);
        float4 c = *(const float4*)(cnt + i);
        float4 o;
        o.x = a.x / fmaxf(c.x, 1.0f);
        o.y = a.y / fmaxf(c.y, 1.0f);
        o.z = a.z / fmaxf(c.z, 1.0f);
        o.w = a.w / fmaxf(c.w, 1.0f);
        *(float4*)(out + i) = o;
        *(float4*)(acc + i) = make_float4(0.f, 0.f, 0.f, 0.f);
    } else {
        for (int j = i; j < n; ++j) {
            out[j] = acc[j] / fmaxf(cnt[j], 1.0f);
            acc[j] = 0.0f;
        }
    }
}

extern "C" void kernel_launch(void* const* d_in, const int* in_sizes, int n_in,
                              void* d_out, int out_size, void* d_ws, size_t ws_size,
                              hipStream_t stream)
{
    const float* topic = (const float*)d_in[0];
    const int*   ei    = (const int*)d_in[1];
    const int*   rei   = (const int*)d_in[2];
    float*       out   = (float*)d_out;

    const int nN = in_sizes[0];
    const int nE = in_sizes[1] / 2;

    float* acc = (float*)d_ws;        // nN floats
    float* cnt = acc + nN;            // nN floats

    const int nblkN = (nN / 4 + 255) / 256;
    const int nblkE = 1024;           // 8192 waves, ~7.6 tiles each

    // acc must start zeroed (d_ws is poisoned by the harness); norm_kernel
    // re-zeroes it after every round, so this is needed only once per call.
    hipMemsetAsync(acc, 0, (size_t)nN * sizeof(float), stream);

    for (int dir = 0; dir < 2; ++dir) {
        const int* s = dir ? rei : ei;
        const int* d = s + nE;
        hipMemsetAsync(cnt, 0, (size_t)nN * sizeof(float), stream);
        const float* hin = topic;
        for (int r = 0; r < 4; ++r) {
            if (r == 0)
                scatter_kernel<true><<<nblkE, BLOCK, 0, stream>>>(s, d, hin, acc, cnt, nE);
            else
                scatter_kernel<false><<<nblkE, BLOCK, 0, stream>>>(s, d, hin, acc, cnt, nE);
            float* o = out + (size_t)(dir * 4 + r) * nN;
            norm_kernel<<<nblkN, 256, 0, stream>>>(acc, cnt, o, nN);
            hin = o;
        }
    }
}